// TATWindowAttention_16492674417482
// MI455X (gfx1250) — compile-verified
//
#include <hip/hip_runtime.h>

typedef float v2f __attribute__((ext_vector_type(2)));
typedef float v8f __attribute__((ext_vector_type(8)));

#define LQ 64
#define DH 32
#define NH 8
#define CDIM 256
#define NWIN 100
#define PADW 65   // LDS row stride for attn tile (64+1 -> conflict-free row walks)
#define LOGIT_MAX 4.605170185988091f

// CDNA5 async copy: global -> LDS, 16 bytes per enabled lane, tracked by ASYNCcnt.
// vdst = LDS byte address (VGPR), vaddr = 32-bit byte offset (VGPR), saddr = SGPR base.
__device__ __forceinline__ void async_copy_b128(unsigned lds_byte_addr,
                                                unsigned gl_byte_off,
                                                const float* sbase) {
    asm volatile("global_load_async_to_lds_b128 %0, %1, %2"
                 :: "v"(lds_byte_addr), "v"(gl_byte_off), "s"(sbase)
                 : "memory");
}

__device__ __forceinline__ void wait_asynccnt0() {
    asm volatile("s_wait_asynccnt 0" ::: "memory");
}

__global__ __launch_bounds__(128) void tat_window_attn_kernel(
    const float* __restrict__ Q, const float* __restrict__ K,
    const float* __restrict__ V, const float* __restrict__ POS,
    const float* __restrict__ LS, float* __restrict__ XOUT,
    float* __restrict__ AOUT)
{
    __shared__ float sQ[LQ * DH];
    __shared__ float sK[LQ * DH];
    __shared__ float sV[LQ * DH];
    __shared__ float sP[LQ * PADW];   // position tile, then raw logits, then probs
    __shared__ float sMax[128];
    __shared__ float sSum[128];

    const int tid  = threadIdx.x;
    const int wave = tid >> 5;
    const int lane = tid & 31;
    const int m    = lane & 15;          // column-in-tile / row-in-A-frag
    const int ksub = (lane >> 4) << 1;   // K sub-offset: 0 for lanes 0-15, 2 for 16-31
    const int rhi  = (lane >> 4) << 3;   // +8 row offset for upper lane half (C/D layout)

    const int gid = blockIdx.x;
    const int h   = gid & (NH - 1);
    const int bn  = gid >> 3;            // b*NWIN + n
    const int b   = bn / NWIN;

    const float* qbase = Q + (size_t)bn * LQ * CDIM + h * DH;
    const float* kbase = K + (size_t)bn * LQ * CDIM + h * DH;
    const float* vbase = V + (size_t)bn * LQ * CDIM + h * DH;
    const float* pbase = POS + (size_t)(b * NH + h) * LQ * LQ;

    // ---- stage Q/K/V head slices via async global->LDS B128 (no VGPR bounce) ----
    const unsigned sQa = (unsigned)(size_t)&sQ[0];
    const unsigned sKa = (unsigned)(size_t)&sK[0];
    const unsigned sVa = (unsigned)(size_t)&sV[0];
    for (int i = tid; i < (LQ * DH) / 4; i += 128) {
        int row = i >> 3;               // 8 float4 per 32-float row
        int f4  = (i & 7) << 2;
        unsigned loff = (unsigned)((row * DH + f4) * 4);
        unsigned goff = (unsigned)((row * CDIM + f4) * 4);
        async_copy_b128(sQa + loff, goff, qbase);
        async_copy_b128(sKa + loff, goff, kbase);
        async_copy_b128(sVa + loff, goff, vbase);
    }
    // ---- stage position tile (scalar, coalesced; padded stride not 16B-aligned) ----
    for (int i = tid; i < LQ * LQ; i += 128) {
        sP[(i >> 6) * PADW + (i & 63)] = pbase[i];
    }

    const float scale = __expf(fminf(LS[h], LOGIT_MAX));

    wait_asynccnt0();      // this wave's async copies have landed in LDS
    __syncthreads();       // ... and every other wave's too

    // ---- GEMM1: S = Q (16x32 per wave) * K^T (32x64), f32 WMMA 16x16x4 ----
    v8f acc[4] = {};
    for (int kk = 0; kk < DH; kk += 4) {
        v2f a;
        a.x = sQ[(wave * 16 + m) * DH + kk + ksub];
        a.y = sQ[(wave * 16 + m) * DH + kk + ksub + 1];
        #pragma unroll
        for (int t = 0; t < 4; ++t) {
            v2f bf;
            bf.x = sK[(t * 16 + m) * DH + kk + ksub];      // B[k][n] = K[n][k]
            bf.y = sK[(t * 16 + m) * DH + kk + ksub + 1];
            acc[t] = __builtin_amdgcn_wmma_f32_16x16x4_f32(
                false, a, false, bf, (short)0, acc[t], false, false);
        }
    }

    // ---- logits = S*scale + position (accumulate into LDS position tile) ----
    #pragma unroll
    for (int t = 0; t < 4; ++t)
        #pragma unroll
        for (int r = 0; r < 8; ++r) {
            int grow = wave * 16 + r + rhi;
            int gcol = t * 16 + m;
            sP[grow * PADW + gcol] = acc[t][r] * scale + sP[grow * PADW + gcol];
        }

    __syncthreads();

    // ---- row softmax: 2 threads per row (half-row each), LDS pair reduction ----
    {
        const int row  = tid >> 1;
        const int half = tid & 1;
        float* rowp = &sP[row * PADW + half * 32];

        float mx = rowp[0];
        #pragma unroll 4
        for (int j = 1; j < 32; ++j) mx = fmaxf(mx, rowp[j]);
        sMax[tid] = mx;
        __syncthreads();
        mx = fmaxf(sMax[row * 2], sMax[row * 2 + 1]);

        float s = 0.f;
        #pragma unroll 4
        for (int j = 0; j < 32; ++j) {
            float e = __expf(rowp[j] - mx);
            rowp[j] = e;
            s += e;
        }
        sSum[tid] = s;
        __syncthreads();
        float inv = 1.f / (sSum[row * 2] + sSum[row * 2 + 1]);

        #pragma unroll 4
        for (int j = 0; j < 32; ++j) rowp[j] *= inv;
    }
    __syncthreads();

    // ---- write attn probabilities (coalesced) ----
    float* aout = AOUT + (size_t)gid * LQ * LQ;   // (B,n,H,L,L) flat == gid major
    for (int i = tid; i < LQ * LQ; i += 128)
        aout[i] = sP[(i >> 6) * PADW + (i & 63)];

    // ---- GEMM2: X = P (16x64 per wave) * V (64x32) ----
    v8f xacc[2] = {};
    for (int kk = 0; kk < LQ; kk += 4) {
        v2f a;
        a.x = sP[(wave * 16 + m) * PADW + kk + ksub];
        a.y = sP[(wave * 16 + m) * PADW + kk + ksub + 1];
        #pragma unroll
        for (int t = 0; t < 2; ++t) {
            v2f bf;
            bf.x = sV[(kk + ksub) * DH + t * 16 + m];      // B[k][n] = V[k][n]
            bf.y = sV[(kk + ksub + 1) * DH + t * 16 + m];
            xacc[t] = __builtin_amdgcn_wmma_f32_16x16x4_f32(
                false, a, false, bf, (short)0, xacc[t], false, false);
        }
    }

    // ---- store X (head slice, 64B contiguous per lane-halfrow) ----
    float* xbase = XOUT + (size_t)bn * LQ * CDIM + h * DH;
    #pragma unroll
    for (int t = 0; t < 2; ++t)
        #pragma unroll
        for (int r = 0; r < 8; ++r) {
            int grow = wave * 16 + r + rhi;
            xbase[grow * CDIM + t * 16 + m] = xacc[t][r];
        }
}

extern "C" void kernel_launch(void* const* d_in, const int* in_sizes, int n_in,
                              void* d_out, int out_size, void* d_ws, size_t ws_size,
                              hipStream_t stream) {
    const float* q   = (const float*)d_in[0];
    const float* k   = (const float*)d_in[1];
    const float* v   = (const float*)d_in[2];
    const float* pos = (const float*)d_in[3];
    const float* ls  = (const float*)d_in[4];

    float* xout = (float*)d_out;
    float* aout = xout + (size_t)16 * NWIN * LQ * CDIM;   // attn follows x

    dim3 grid(16 * NWIN * NH);   // one workgroup per (b, n, h)
    tat_window_attn_kernel<<<grid, 128, 0, stream>>>(q, k, v, pos, ls, xout, aout);
}